// Inter_sub_sim_48756468744578
// MI455X (gfx1250) — compile-verified
//
#include <hip/hip_runtime.h>

// Problem constants (match reference)
#define BB    32768
#define TWOB  65536
#define DD    512
#define KK    256
#define LL    8

typedef __attribute__((ext_vector_type(2))) float v2f;
typedef __attribute__((ext_vector_type(4))) float v4f;
typedef __attribute__((ext_vector_type(8))) float v8f;

// feat row gr in [0, 2B): gr<B -> X[gr,0,:], else X[gr-B,1,:]
__device__ __forceinline__ const float* feat_row(const float* __restrict__ X, int gr) {
    int b = (gr < BB) ? gr : (gr - BB);
    int v = (gr < BB) ? 0 : 1;
    return X + ((size_t)b * 2 + v) * DD;
}

__device__ __forceinline__ int row_gid(const int* __restrict__ sub, const int* __restrict__ lab, int gr) {
    int rr = (gr < BB) ? gr : (gr - BB);
    return sub[rr] * LL + lab[rr];
}

// ---------------- K1: zero workspace + output ----------------
__global__ void __launch_bounds__(256)
k_zero(float* __restrict__ ws, int n, float* __restrict__ out) {
    int i = blockIdx.x * 256 + threadIdx.x;
    if (i < n) ws[i] = 0.0f;
    if (blockIdx.x == 0 && threadIdx.x == 0) out[0] = 0.0f;
}

// ---------------- K2a: group counts (LDS aggregated) ----------------
__global__ void __launch_bounds__(256)
k_counts(const int* __restrict__ sub, const int* __restrict__ lab, float* __restrict__ counts) {
    __shared__ float c[KK];
    int t = threadIdx.x;
    c[t] = 0.0f;
    __syncthreads();
    int base = blockIdx.x * 1024;
    for (int r = base + t; r < base + 1024; r += 256)
        atomicAdd(&c[row_gid(sub, lab, r)], 1.0f);
    __syncthreads();
    atomicAdd(&counts[t], c[t]);
}

// ---------------- K2b: group feature sums (LDS [256 x 64] tile) ----------------
__global__ void __launch_bounds__(256)
k_sums(const float* __restrict__ X, const int* __restrict__ sub, const int* __restrict__ lab,
       float* __restrict__ sums) {
    __shared__ float tile[KK * 64];   // 64 KB
    int t = threadIdx.x;
    int dchunk = blockIdx.x;          // 0..7
    int rowBase = blockIdx.y * 2048;  // 32 y-blocks
    for (int i = t; i < KK * 64; i += 256) tile[i] = 0.0f;
    __syncthreads();
    int dl = t & 63;
    int rsub = t >> 6;                // 4 rows in flight
    for (int i = 0; i < 2048; i += 4) {
        int gr  = rowBase + i + rsub;
        int gid = row_gid(sub, lab, gr);
        float v = feat_row(X, gr)[dchunk * 64 + dl];
        atomicAdd(&tile[gid * 64 + dl], v);
    }
    __syncthreads();
    for (int i = t; i < KK * 64; i += 256) {
        int g = i >> 6, d = i & 63;
        atomicAdd(&sums[(size_t)g * DD + dchunk * 64 + d], tile[i]);
    }
}

// ---------------- K3: centroid = sums / counts (in place) ----------------
__global__ void __launch_bounds__(256)
k_centroid(float* __restrict__ sums, const float* __restrict__ counts) {
    int i = blockIdx.x * 256 + threadIdx.x;   // 512 blocks -> 131072
    int g = i >> 9;
    float c = counts[g];
    sums[i] = (c > 0.0f) ? sums[i] / c : 0.0f;
}

// ---------------- K4: per-row distance, accumulate sqrt(dist) per group ----------------
__global__ void __launch_bounds__(256)
k_dist(const float* __restrict__ X, const int* __restrict__ sub, const int* __restrict__ lab,
       const float* __restrict__ centroid, float* __restrict__ sqrtsum) {
    int t = threadIdx.x;
    int lane = t & 31;
    int gr = blockIdx.x * 8 + (t >> 5);       // wave per row
    int gid = row_gid(sub, lab, gr);
    const v4f* fr = (const v4f*)feat_row(X, gr);
    const v4f* cr = (const v4f*)(centroid + (size_t)gid * DD);
    float ss = 0.0f;
    #pragma unroll
    for (int k = 0; k < 4; ++k) {
        int i4 = k * 32 + lane;
        v4f f = fr[i4];
        v4f c = cr[i4];
        float d0 = f.x - c.x, d1 = f.y - c.y, d2 = f.z - c.z, d3 = f.w - c.w;
        ss += d0 * d0 + d1 * d1 + d2 * d2 + d3 * d3;
    }
    #pragma unroll
    for (int off = 16; off > 0; off >>= 1) ss += __shfl_xor(ss, off, 32);
    if (lane == 0) atomicAdd(&sqrtsum[gid], sqrtf(sqrtf(ss)));  // ss^0.25 = sqrt(dist)
}

// ---------------- K5: density (single block: max, bitonic sort for quantiles, mean) ----------------
__global__ void __launch_bounds__(256)
k_density(const float* __restrict__ counts, const float* __restrict__ sqrtsum,
          float* __restrict__ density) {
    __shared__ float s[KK];
    __shared__ float red[KK];
    int t = threadIdx.x;
    float cnt = counts[t];
    bool valid = cnt > 1.0f;
    float dens = valid ? (sqrtsum[t] / cnt) / logf(cnt + 10.0f) : -3.4e38f;
    red[t] = dens;
    __syncthreads();
    for (int off = 128; off > 0; off >>= 1) {
        if (t < off) red[t] = fmaxf(red[t], red[t + off]);
        __syncthreads();
    }
    float dmax = red[0];
    __syncthreads();
    float d0 = valid ? dens : dmax;
    s[t] = d0;
    __syncthreads();
    // bitonic sort ascending over 256 values
    for (int ksz = 2; ksz <= 256; ksz <<= 1) {
        for (int j = ksz >> 1; j > 0; j >>= 1) {
            int ixj = t ^ j;
            if (ixj > t) {
                float a = s[t], b = s[ixj];
                bool up = ((t & ksz) == 0);
                if ((a > b) == up) { s[t] = b; s[ixj] = a; }
            }
            __syncthreads();
        }
    }
    float q10 = 0.5f * (s[25] + s[26]);     // (n-1)*0.1 = 25.5, linear interp
    float q90 = 0.5f * (s[229] + s[230]);   // (n-1)*0.9 = 229.5
    d0 = fminf(fmaxf(d0, q10), q90);
    red[t] = d0;
    __syncthreads();
    for (int off = 128; off > 0; off >>= 1) {
        if (t < off) red[t] += red[t + off];
        __syncthreads();
    }
    float mean = red[0] * (1.0f / 256.0f);
    density[t] = 0.1f * d0 / mean;
}

// ---------------- K6: fused WMMA GEMM (feat @ centroid^T / density) + masked CE loss ----------------
// Block = 256 threads (8 waves); 64-row stripe x all 256 columns.
// Wave w owns cols [w*32, w*32+32) (2 n-tiles), loops 4 m-tiles -> 8 WMMA accumulators.
// Dynamic LDS: 131072 B = A stripe [64][512] f32, later reused as sim [64][257].
__global__ void __launch_bounds__(256)
k_gemm_loss(const float* __restrict__ X, const int* __restrict__ sub, const int* __restrict__ lab,
            const float* __restrict__ centroid, const float* __restrict__ density,
            float* __restrict__ out) {
    extern __shared__ float smem[];   // 32768 floats
    const int t = threadIdx.x;
    const int lane = t & 31;
    const int wave = t >> 5;
    const int r0 = blockIdx.x * 64;

    // stage A stripe (coalesced float4): 64 rows x 128 float4
    for (int i = t; i < 64 * 128; i += 256) {
        int row = i >> 7, d4 = (i & 127) << 2;
        const float* rp = feat_row(X, r0 + row);
        *(v4f*)(&smem[row * DD + d4]) = *(const v4f*)(rp + d4);
    }
    __syncthreads();

    const int half  = lane >> 4;     // 0: K=k,k+1  1: K=k+2,k+3 (ISA 16x4 f32 A layout)
    const int sub16 = lane & 15;     // M (for A) / N (for B)

    v8f acc[4][2];
    #pragma unroll
    for (int mt = 0; mt < 4; ++mt)
        #pragma unroll
        for (int nt = 0; nt < 2; ++nt)
            acc[mt][nt] = {};

    const float* aptr[4];
    #pragma unroll
    for (int mt = 0; mt < 4; ++mt)
        aptr[mt] = &smem[(mt * 16 + sub16) * DD + half * 2];
    const float* bptr0 = centroid + (size_t)(wave * 32 + 0  + sub16) * DD + half * 2;
    const float* bptr1 = centroid + (size_t)(wave * 32 + 16 + sub16) * DD + half * 2;

    for (int k = 0; k < DD; k += 4) {
        v2f b0 = *(const v2f*)(bptr0 + k);
        v2f b1 = *(const v2f*)(bptr1 + k);
        #pragma unroll
        for (int mt = 0; mt < 4; ++mt) {
            v2f a = *(const v2f*)(aptr[mt] + k);
            acc[mt][0] = __builtin_amdgcn_wmma_f32_16x16x4_f32(false, a, false, b0, (short)0, acc[mt][0], false, false);
            acc[mt][1] = __builtin_amdgcn_wmma_f32_16x16x4_f32(false, a, false, b1, (short)0, acc[mt][1], false, false);
        }
    }
    __syncthreads();   // all waves done reading A; reuse smem for sim [64][257]

    {   // C/D layout: VGPR j -> M = j (lanes 0-15) / 8+j (lanes 16-31); N = lane&15
        int col0 = wave * 32 + sub16;
        int col1 = col0 + 16;
        float sc0 = 1.0f / density[col0];
        float sc1 = 1.0f / density[col1];
        int rbase = half * 8;
        #pragma unroll
        for (int mt = 0; mt < 4; ++mt) {
            #pragma unroll
            for (int j = 0; j < 8; ++j) {
                int r = mt * 16 + rbase + j;
                smem[r * 257 + col0] = acc[mt][0][j] * sc0;
                smem[r * 257 + col1] = acc[mt][1][j] * sc1;
            }
        }
    }
    __syncthreads();

    // per-row masked log-softmax CE (thread t owns row t)
    if (t < 64) {
        float* rowp = &smem[t * 257];
        int gr = r0 + t;
        int rr = (gr < BB) ? gr : (gr - BB);
        int sj = sub[rr], lb = lab[rr];
        float rmax = -3.4e38f;
        for (int c = 0; c < KK; ++c) rmax = fmaxf(rmax, rowp[c]);
        float pmax = -3.4e38f, msum = 0.0f, spsum = 0.0f;
        for (int c = 0; c < KK; ++c) {
            bool m = (sj != (c >> 3)) && (lb == (c & 7));
            float p = m ? (rowp[c] - rmax) : 0.0f;   // positive_sim = (sim - max) * mask
            rowp[c] = p;
            pmax = fmaxf(pmax, p);
            if (m) { msum += 1.0f; spsum += p; }
        }
        float se = 0.0f;
        for (int c = 0; c < KK; ++c) se += __expf(rowp[c] - pmax);
        // -sum(mask*logp)/msum = pmax + log(se) - spsum/msum
        float loss = pmax + __logf(se) - spsum / msum;
        atomicAdd(out, loss * (1.0f / (float)TWOB));
    }
}

extern "C" void kernel_launch(void* const* d_in, const int* in_sizes, int n_in,
                              void* d_out, int out_size, void* d_ws, size_t ws_size,
                              hipStream_t stream) {
    const float* X   = (const float*)d_in[0];   // [B, 2, 512] f32
    const int*   sub = (const int*)d_in[1];     // [B]
    const int*   lab = (const int*)d_in[2];     // [B]
    float* out = (float*)d_out;                 // scalar loss

    float* ws      = (float*)d_ws;
    float* sums    = ws;                 // 131072 (becomes centroid)
    float* counts  = ws + 131072;        // 256
    float* sqrts   = ws + 131328;        // 256
    float* density = ws + 131584;        // 256
    const int nzero = 131840;

    // opt-in for 128 KB dynamic LDS (no-op if driver already allows it)
    (void)hipFuncSetAttribute((const void*)k_gemm_loss,
                              hipFuncAttributeMaxDynamicSharedMemorySize, 131072);

    k_zero    <<<(nzero + 255) / 256, 256, 0, stream>>>(ws, nzero, out);
    k_counts  <<<64, 256, 0, stream>>>(sub, lab, counts);
    k_sums    <<<dim3(8, 32), 256, 0, stream>>>(X, sub, lab, sums);
    k_centroid<<<512, 256, 0, stream>>>(sums, counts);
    k_dist    <<<TWOB / 8, 256, 0, stream>>>(X, sub, lab, sums, sqrts);
    k_density <<<1, 256, 0, stream>>>(counts, sqrts, density);
    k_gemm_loss<<<TWOB / 64, 256, 131072, stream>>>(X, sub, lab, sums, density, out);
}